// Lcat_Layer_18605798326370
// MI455X (gfx1250) — compile-verified
//
#include <hip/hip_runtime.h>
#include <hip/hip_bf16.h>
#include <math.h>

// -------------------------------------------------------------------------
// KG entity encoder layer for MI455X (gfx1250, wave32).
//   - transformer block degenerates: softmax over len-1 seq == 1 => ctx == v
//     (exact), so wq/wk are dead. 5 surviving 128x128 GEMMs use
//     V_WMMA_F32_16X16X4_F32 (full-f32 matrix pipe, validation-exact math;
//     GEMMs are not the bottleneck so no reason to downcast).
//   - weights staged in LDS *transposed + bank-XOR-swizzled* so each WMMA
//     B fragment is one aligned ds_load_b64 (no VGPR repacking).
//   - edge phases (GCN/GAT scatter) are the bandwidth bound: float4 gathers +
//     global f32 atomics.
// -------------------------------------------------------------------------

typedef float v2f __attribute__((ext_vector_type(2)));
typedef float v8f __attribute__((ext_vector_type(8)));

#define D128 128

// ---------------- LayerNorm: one wave per row (32 lanes x float4) ----------
__global__ void ln128_kernel(const float* __restrict__ X,
                             const float* __restrict__ g,
                             const float* __restrict__ b,
                             float* __restrict__ Y, int nrows) {
  int wave = (int)((blockIdx.x * blockDim.x + threadIdx.x) >> 5);
  int lane = threadIdx.x & 31;
  if (wave >= nrows) return;
  const float* x = X + (size_t)wave * D128;
  float4 v = *(const float4*)(x + lane * 4);
  float s = v.x + v.y + v.z + v.w;
  #pragma unroll
  for (int off = 16; off; off >>= 1) s += __shfl_xor(s, off, 32);
  float mu = s * (1.0f / 128.0f);
  float dx = v.x - mu, dy = v.y - mu, dz = v.z - mu, dw = v.w - mu;
  float q = dx * dx + dy * dy + dz * dz + dw * dw;
  #pragma unroll
  for (int off = 16; off; off >>= 1) q += __shfl_xor(q, off, 32);
  float rs = rsqrtf(q * (1.0f / 128.0f) + 1e-5f);
  float4 gg = *(const float4*)(g + lane * 4);
  float4 bb = *(const float4*)(b + lane * 4);
  float4 o;
  o.x = dx * rs * gg.x + bb.x;
  o.y = dy * rs * gg.y + bb.y;
  o.z = dz * rs * gg.z + bb.z;
  o.w = dw * rs * gg.w + bb.w;
  *(float4*)(Y + (size_t)wave * D128 + lane * 4) = o;
}

// ---------------- GEMM: out[r,n] = ep( X[r,:]@W[:,n] + bias[n] ) -----------
// X: [nrows,128] row-major, W: [128,128] row-major (k-major, numpy @).
// Each wave -> 16x128 strip: 8 accum tiles, K swept 4 at a time with
// v_wmma_f32_16x16x4_f32.
// W staged in LDS transposed with XOR bank swizzle:
//   Wt[dword n*128 + (k ^ ((n&15)*4))] = W[k][n]
// => B fragment (W[kk][n], W[kk+1][n]) is ONE aligned 8-byte ds load
//    (pair stays contiguous: swizzle const is mult-of-4, kk is even), and the
//    16 lanes of each half-wave hit 16 distinct banks.
// EP: 0 = none, 1 = add residual, 2 = exact GELU.
template <int EP>
__global__ void gemm128_kernel(const float* __restrict__ X,
                               const float* __restrict__ W,
                               const float* __restrict__ bias,
                               const float* __restrict__ res,
                               float* __restrict__ out, int nrows) {
  __shared__ float Wt[D128 * D128];  // 64 KB
  // cooperative transpose+swizzle stage (coalesced float4 reads)
  for (int i = threadIdx.x; i < D128 * D128 / 4; i += blockDim.x) {
    int k  = i >> 5;           // row of W
    int n4 = (i & 31) * 4;     // col group
    float4 w = *(const float4*)&W[k * D128 + n4];
    Wt[(n4 + 0) * D128 + (k ^ (((n4 + 0) & 15) * 4))] = w.x;
    Wt[(n4 + 1) * D128 + (k ^ (((n4 + 1) & 15) * 4))] = w.y;
    Wt[(n4 + 2) * D128 + (k ^ (((n4 + 2) & 15) * 4))] = w.z;
    Wt[(n4 + 3) * D128 + (k ^ (((n4 + 3) & 15) * 4))] = w.w;
  }
  __syncthreads();

  int wave = threadIdx.x >> 5;
  int lane = threadIdx.x & 31;
  int r0 = (blockIdx.x * 8 + wave) * 16;  // 16-row tile owned by this wave
  if (r0 >= nrows) return;                 // wave-uniform guard (EXEC stays full)

  int m  = lane & 15;   // A-row / B-col within tile
  int kh = lane >> 4;   // K half-select per ISA A/B fragment layout
  int swz   = m * 4;    // bank swizzle constant for this lane's column
  int laneB = 2 * kh;   // K offset of this lane half

  v8f acc[8] = {};      // 8 tiles x 16x16 f32 accumulators
  const float* xrow = X + (size_t)(r0 + m) * D128 + laneB;
  // per-lane LDS base (dwords): column block + lane-half K offset
  const float* wt0 = &Wt[m * D128 + laneB];

  for (int k0 = 0; k0 < D128; k0 += 4) {
    v2f a = *(const v2f*)(xrow + k0);   // A frag: (X[m][kk], X[m][kk+1])
    int kx = k0 ^ swz;                  // swizzled K index (mult of 4)
    #pragma unroll
    for (int t = 0; t < 8; ++t) {
      // B frag: (W[kk][n], W[kk+1][n]) -> single aligned ds_load_b64
      v2f bfrag = *(const v2f*)(wt0 + t * 16 * D128 + kx);
      acc[t] = __builtin_amdgcn_wmma_f32_16x16x4_f32(
          /*neg_a=*/false, a, /*neg_b=*/false, bfrag,
          /*c_mod=*/(short)0, acc[t], /*reuse_a=*/false, /*reuse_b=*/false);
    }
  }

  // epilogue + store. C/D layout: VGPR i -> row (i + 8*kh), col n = t*16 + m.
  #pragma unroll
  for (int t = 0; t < 8; ++t) {
    int n = t * 16 + m;
    float bn = bias[n];
    #pragma unroll
    for (int i = 0; i < 8; ++i) {
      int row = r0 + i + 8 * kh;
      float v = acc[t][i] + bn;
      if (EP == 2) v = 0.5f * v * (1.0f + erff(v * 0.70710678118654752f));
      if (EP == 1) v += res[(size_t)row * D128 + n];
      out[(size_t)row * D128 + n] = v;
    }
  }
}

// ---------------- GCN: scatter edge_weight * gcn_in[col] into row ----------
__global__ void gcn_scatter_kernel(const int* __restrict__ ei,
                                   const float* __restrict__ ew,
                                   const float* __restrict__ gin,
                                   float* __restrict__ neigh,
                                   float* __restrict__ rowsum, int E_) {
  int e = (int)((blockIdx.x * blockDim.x + threadIdx.x) >> 5);
  int lane = threadIdx.x & 31;
  if (e >= E_) return;
  int row = ei[e];
  int col = ei[E_ + e];
  float w = ew[e];
  if (lane == 0) atomicAdd(rowsum + row, w);
  float4 v = *(const float4*)(gin + (size_t)col * D128 + lane * 4);
  float* dst = neigh + (size_t)row * D128 + lane * 4;
  atomicAdd(dst + 0, w * v.x);
  atomicAdd(dst + 1, w * v.y);
  atomicAdd(dst + 2, w * v.z);
  atomicAdd(dst + 3, w * v.w);
}

__global__ void gcn_combine_kernel(const float* __restrict__ gin,
                                   const float* __restrict__ neigh,
                                   const float* __restrict__ rowsum,
                                   const float* __restrict__ lmbda,
                                   float* __restrict__ outp, int total) {
  int idx = blockIdx.x * blockDim.x + threadIdx.x;
  if (idx >= total) return;
  int r = idx >> 7;
  float l1 = lmbda[0], l2 = lmbda[1];
  float denom = 1.0f + l2 * rowsum[r];
  outp[idx] = (gin[idx] + l2 * neigh[idx]) / denom * l1;
}

// ---------------- GAT: per-node attention scalars s_l, s_r -----------------
__global__ void gat_scores_kernel(const float* __restrict__ gat_in,
                                  const float* __restrict__ watt,  // [256]
                                  const int* __restrict__ batch_ids,
                                  float* __restrict__ s_l,
                                  float* __restrict__ s_r, int nrows) {
  int wave = (int)((blockIdx.x * blockDim.x + threadIdx.x) >> 5);
  int lane = threadIdx.x & 31;
  if (wave >= nrows) return;
  int nl = batch_ids[wave];
  float4 a  = *(const float4*)(gat_in + (size_t)nl * D128 + lane * 4);
  float4 wl = *(const float4*)(watt + lane * 4);
  float4 b  = *(const float4*)(gat_in + (size_t)wave * D128 + lane * 4);
  float4 wr = *(const float4*)(watt + D128 + lane * 4);
  float sl = a.x * wl.x + a.y * wl.y + a.z * wl.z + a.w * wl.w;
  float sr = b.x * wr.x + b.y * wr.y + b.z * wr.z + b.w * wr.w;
  #pragma unroll
  for (int off = 16; off; off >>= 1) {
    sl += __shfl_xor(sl, off, 32);
    sr += __shfl_xor(sr, off, 32);
  }
  if (lane == 0) { s_l[wave] = sl; s_r[wave] = sr; }
}

// ---------------- GAT: per-edge attention scatter --------------------------
__global__ void gat_scatter_kernel(const int* __restrict__ bei,
                                   const float* __restrict__ s_l,
                                   const float* __restrict__ s_r,
                                   const float* __restrict__ gat_in,
                                   float* __restrict__ e_out,
                                   float* __restrict__ e_rowsum, int E2) {
  int e = (int)((blockIdx.x * blockDim.x + threadIdx.x) >> 5);
  int lane = threadIdx.x & 31;
  if (e >= E2) return;
  int brow = bei[e];
  int bcol = bei[E2 + e];
  float ev = s_l[brow] + s_r[bcol];
  float lr = ev > 0.0f ? ev : 0.2f * ev;   // leaky_relu(e, 0.2)
  float att = expf(-lr);
  if (lane == 0) atomicAdd(e_rowsum + brow, att);
  float4 v = *(const float4*)(gat_in + (size_t)bcol * D128 + lane * 4);
  float* dst = e_out + (size_t)brow * D128 + lane * 4;
  atomicAdd(dst + 0, att * v.x);
  atomicAdd(dst + 1, att * v.y);
  atomicAdd(dst + 2, att * v.z);
  atomicAdd(dst + 3, att * v.w);
}

__global__ void gat_final_kernel(float* __restrict__ outp,
                                 const float* __restrict__ e_rowsum,
                                 const float* __restrict__ prelu_a, int total) {
  int idx = blockIdx.x * blockDim.x + threadIdx.x;
  if (idx >= total) return;
  int r = idx >> 7;
  float v = outp[idx] / e_rowsum[r];
  float a = prelu_a[0];
  outp[idx] = v > 0.0f ? v : a * v;
}

// -------------------------------------------------------------------------
extern "C" void kernel_launch(void* const* d_in, const int* in_sizes, int n_in,
                              void* d_out, int out_size, void* d_ws, size_t ws_size,
                              hipStream_t stream) {
  const float* ent     = (const float*)d_in[0];
  const float* ew      = (const float*)d_in[1];
  const float* ln1_g   = (const float*)d_in[2];
  const float* ln1_b   = (const float*)d_in[3];
  // d_in[4..7] = wq,bq,wk,bk : dead (softmax over length-1 axis == 1 => ctx == v)
  const float* wv      = (const float*)d_in[8];
  const float* bv      = (const float*)d_in[9];
  const float* wo      = (const float*)d_in[10];
  const float* bo      = (const float*)d_in[11];
  const float* ln2_g   = (const float*)d_in[12];
  const float* ln2_b   = (const float*)d_in[13];
  const float* w1      = (const float*)d_in[14];
  const float* b1      = (const float*)d_in[15];
  const float* w2      = (const float*)d_in[16];
  const float* b2      = (const float*)d_in[17];
  const float* lmbda   = (const float*)d_in[18];
  const float* gat_w   = (const float*)d_in[19];
  const float* gat_b   = (const float*)d_in[20];
  const float* watt    = (const float*)d_in[21];  // [256]
  const float* prelu_a = (const float*)d_in[22];
  const int*   ei      = (const int*)d_in[23];    // [2,E]
  const int*   bids    = (const int*)d_in[24];    // [N]
  const int*   bei     = (const int*)d_in[25];    // [2,E2]

  const int N_  = in_sizes[0] / D128;   // 100000 (multiple of 16)
  const int E_  = in_sizes[1];          // 1200000
  const int E2  = in_sizes[25] / 2;     // 1300000
  float* outp   = (float*)d_out;

  // workspace: 3 big [N,128] buffers + 4 [N] scalar arrays
  size_t big = (size_t)N_ * D128 * sizeof(float);
  char*  ws  = (char*)d_ws;
  float* A        = (float*)(ws);
  float* B        = (float*)(ws + big);
  float* C        = (float*)(ws + 2 * big);
  float* rowsum   = (float*)(ws + 3 * big);
  float* e_rowsum = rowsum + N_;
  float* s_l      = e_rowsum + N_;  // fully written before read
  float* s_r      = s_l + N_;

  const int TPB = 256;
  int lnBlocks   = (N_ + 7) / 8;                 // 8 waves/block, wave per row
  int rowTiles   = N_ / 16;                      // 6250
  int gemmBlocks = (rowTiles + 7) / 8;           // 8 waves/block, wave per tile
  int elemBlocks = (N_ * D128 + TPB - 1) / TPB;
  int gcnBlocks  = (E_ + 7) / 8;
  int gatBlocks  = (E2 + 7) / 8;

  // ---- transformer block (ctx == v exactly) ----
  ln128_kernel<<<lnBlocks, TPB, 0, stream>>>(ent, ln1_g, ln1_b, A, N_);      // A = y
  gemm128_kernel<0><<<gemmBlocks, TPB, 0, stream>>>(A, wv, bv, nullptr, B, N_); // B = v
  gemm128_kernel<1><<<gemmBlocks, TPB, 0, stream>>>(B, wo, bo, ent, A, N_);     // A = x
  ln128_kernel<<<lnBlocks, TPB, 0, stream>>>(A, ln2_g, ln2_b, B, N_);        // B = y2
  gemm128_kernel<2><<<gemmBlocks, TPB, 0, stream>>>(B, w1, b1, nullptr, C, N_); // C = gelu(...)
  gemm128_kernel<1><<<gemmBlocks, TPB, 0, stream>>>(C, w2, b2, A, B, N_);       // B = gcn_in

  // ---- GCN ----
  hipMemsetAsync(C, 0, big, stream);                                  // neigh_sum
  hipMemsetAsync(rowsum, 0, 2 * (size_t)N_ * sizeof(float), stream);  // rowsum + e_rowsum
  hipMemsetAsync(d_out, 0, (size_t)out_size * sizeof(float), stream); // e_out accum
  gcn_scatter_kernel<<<gcnBlocks, TPB, 0, stream>>>(ei, ew, B, C, rowsum, E_);
  gcn_combine_kernel<<<elemBlocks, TPB, 0, stream>>>(B, C, rowsum, lmbda, A, N_ * D128); // A = gcn_out

  // ---- GAT ----
  gemm128_kernel<0><<<gemmBlocks, TPB, 0, stream>>>(A, gat_w, gat_b, nullptr, C, N_);   // C = gat_in
  gat_scores_kernel<<<lnBlocks, TPB, 0, stream>>>(C, watt, bids, s_l, s_r, N_);
  gat_scatter_kernel<<<gatBlocks, TPB, 0, stream>>>(bei, s_l, s_r, C, outp, e_rowsum, E2);
  gat_final_kernel<<<elemBlocks, TPB, 0, stream>>>(outp, e_rowsum, prelu_a, N_ * D128);
}